// RNNTLoss_87385404604556
// MI455X (gfx1250) — compile-verified
//
#include <hip/hip_runtime.h>
#include <math.h>
#include <stdint.h>

typedef float v4f __attribute__((ext_vector_type(4)));

// Problem constants (match reference).
#define B_ 8
#define T_ 200
#define U_ 50
#define V_ 1024
#define ROWS (B_ * T_ * (U_ + 1))   // 81600 softmax rows
#define NB   (B_ * T_ * (U_ + 1))   // lp_blank elements
#define NE   (B_ * T_ * U_)         // lp_emit elements

// ---------------------------------------------------------------------------
// Kernel 1: per-row logsumexp over V=1024 + gather blank/label log-probs.
// One wave32 per row; 8 waves (256 threads) per block; 32KB LDS staging.
// HBM -> LDS via CDNA5 async-to-LDS (ASYNCcnt); LDS -> VGPR via asm ds_load
// so the optimizer cannot fold the reads of asm-written LDS.
// ---------------------------------------------------------------------------
__global__ __launch_bounds__(256) void rnnt_lse_gather_kernel(
    const float* __restrict__ acts,     // (B,T,U+1,V)
    const int*   __restrict__ labels,   // (B,U)
    float*       __restrict__ lp_blank, // (B,T,U+1)
    float*       __restrict__ lp_emit)  // (B,T,U)
{
    __shared__ __align__(16) float smem[8 * V_];  // 8 rows x 4KB

    const int lane = threadIdx.x & 31;
    const int wave = threadIdx.x >> 5;
    const int row  = blockIdx.x * 8 + wave;       // ROWS % 8 == 0
    if (row >= ROWS) return;

    // Row coordinates + early (latency-hidden) label fetch.
    const int u   = row % (U_ + 1);
    const int bt  = row / (U_ + 1);               // b*T + t
    const int b   = bt / T_;
    const int lab = (u < U_) ? labels[b * U_ + u] : 0;

    // DS byte offset of smem = low 32 bits of its generic address.
    const unsigned sbase    = (unsigned)(uintptr_t)(const void*)smem;
    const unsigned row_lds  = sbase + (unsigned)(wave * (V_ * 4));
    const unsigned lds_addr = row_lds + (unsigned)(lane * 16);
    const float*   gaddr    = acts + (size_t)row * V_ + lane * 4;

    // --- async copy this wave's 4KB row into LDS (inst offset is added to
    //     BOTH the global and the LDS address) ---
    asm volatile(
        "global_load_async_to_lds_b128 %0, %1, off\n\t"
        "global_load_async_to_lds_b128 %0, %1, off offset:512\n\t"
        "global_load_async_to_lds_b128 %0, %1, off offset:1024\n\t"
        "global_load_async_to_lds_b128 %0, %1, off offset:1536\n\t"
        "global_load_async_to_lds_b128 %0, %1, off offset:2048\n\t"
        "global_load_async_to_lds_b128 %0, %1, off offset:2560\n\t"
        "global_load_async_to_lds_b128 %0, %1, off offset:3072\n\t"
        "global_load_async_to_lds_b128 %0, %1, off offset:3584"
        :
        : "v"(lds_addr), "v"(gaddr)
        : "memory");
    asm volatile("s_wait_asynccnt 0" ::: "memory");

    // --- read the row back: lane reads float4 indices lane + 32*i ---
    v4f A[8];
    asm volatile(
        "ds_load_b128 %0, %8\n\t"
        "ds_load_b128 %1, %8 offset:512\n\t"
        "ds_load_b128 %2, %8 offset:1024\n\t"
        "ds_load_b128 %3, %8 offset:1536\n\t"
        "ds_load_b128 %4, %8 offset:2048\n\t"
        "ds_load_b128 %5, %8 offset:2560\n\t"
        "ds_load_b128 %6, %8 offset:3072\n\t"
        "ds_load_b128 %7, %8 offset:3584\n\t"
        "s_wait_dscnt 0"
        : "=v"(A[0]), "=v"(A[1]), "=v"(A[2]), "=v"(A[3]),
          "=v"(A[4]), "=v"(A[5]), "=v"(A[6]), "=v"(A[7])
        : "v"(lds_addr)
        : "memory");

    // --- pass 1: row max (registers) ---
    float m = -INFINITY;
#pragma unroll
    for (int i = 0; i < 8; ++i) {
        m = fmaxf(m, fmaxf(fmaxf(A[i].x, A[i].y), fmaxf(A[i].z, A[i].w)));
    }
#pragma unroll
    for (int off = 16; off > 0; off >>= 1)
        m = fmaxf(m, __shfl_xor(m, off, 32));

    // --- pass 2: sum of exp(x - m) (registers) ---
    float s = 0.0f;
#pragma unroll
    for (int i = 0; i < 8; ++i) {
        s += __expf(A[i].x - m) + __expf(A[i].y - m) +
             __expf(A[i].z - m) + __expf(A[i].w - m);
    }
#pragma unroll
    for (int off = 16; off > 0; off >>= 1)
        s += __shfl_xor(s, off, 32);

    const float lse = m + __logf(s);

    // --- dynamic gathers from LDS (blank = elem 0, label = elem lab) ---
    float e_blank, e_lab;
    const unsigned gl = row_lds + (unsigned)(lab * 4);
    asm volatile(
        "ds_load_b32 %0, %2\n\t"
        "ds_load_b32 %1, %3\n\t"
        "s_wait_dscnt 0"
        : "=v"(e_blank), "=v"(e_lab)
        : "v"(row_lds), "v"(gl)
        : "memory");

    if (lane == 0) {
        lp_blank[row] = e_blank - lse;
        if (u < U_) lp_emit[bt * U_ + u] = e_lab - lse;
    }
}

// ---------------------------------------------------------------------------
// Kernel 2: alpha wavefront recursion per batch + final loss accumulation.
// alpha[t][u] = logaddexp(alpha[t-1][u] + blank[t-1][u],
//                         alpha[t][u-1] + emit[t][u-1])
// Diagonal d = t+u depends only on diagonal d-1 -> LDS ping-pong buffers.
// ---------------------------------------------------------------------------
__global__ __launch_bounds__(64) void rnnt_alpha_kernel(
    const float* __restrict__ lp_blank,
    const float* __restrict__ lp_emit,
    const int*   __restrict__ act_lens,
    const int*   __restrict__ label_lens,
    float*       __restrict__ out)       // scalar, pre-zeroed
{
    __shared__ float dbuf[2][U_ + 1];

    const int b  = blockIdx.x;
    const int u  = threadIdx.x;          // 0..63 (only 0..U_ active)
    const int tl = act_lens[b] - 1;
    const int ul = label_lens[b];
    const float* blank_b = lp_blank + (size_t)b * T_ * (U_ + 1);
    const float* emit_b  = lp_emit  + (size_t)b * T_ * U_;

    int cur = 0;
    for (int d = 0; d <= (T_ - 1) + U_; ++d) {
        const bool valid = (u <= U_) && (u <= d) && ((d - u) <= (T_ - 1));
        if (valid) {
            const int t = d - u;
            float val;
            if (t == 0 && u == 0) {
                val = 0.0f;
            } else if (t == 0) {
                val = dbuf[cur ^ 1][u - 1] + emit_b[u - 1];          // cumsum row
            } else if (u == 0) {
                val = dbuf[cur ^ 1][0] + blank_b[(t - 1) * (U_ + 1)];
            } else {
                const float a = dbuf[cur ^ 1][u]     + blank_b[(t - 1) * (U_ + 1) + u];
                const float c = dbuf[cur ^ 1][u - 1] + emit_b[t * U_ + (u - 1)];
                const float mx = fmaxf(a, c);
                val = mx + log1pf(__expf(-fabsf(a - c)));            // logaddexp
            }
            dbuf[cur][u] = val;
            if (t == tl && u == ul) {
                const float ll = val + blank_b[tl * (U_ + 1) + ul];
                atomicAdd(out, -ll);                                  // loss = -sum ll
            }
        }
        __syncthreads();
        cur ^= 1;
    }
}

// ---------------------------------------------------------------------------
extern "C" void kernel_launch(void* const* d_in, const int* in_sizes, int n_in,
                              void* d_out, int out_size, void* d_ws, size_t ws_size,
                              hipStream_t stream) {
    const float* acts       = (const float*)d_in[0];
    const int*   labels     = (const int*)  d_in[1];
    const int*   act_lens   = (const int*)  d_in[2];
    const int*   label_lens = (const int*)  d_in[3];
    float*       out        = (float*)d_out;

    float* lp_blank = (float*)d_ws;          // NB floats
    float* lp_emit  = lp_blank + NB;         // NE floats  (~646KB total)

    hipMemsetAsync(out, 0, sizeof(float), stream);  // graph-capture safe

    rnnt_lse_gather_kernel<<<ROWS / 8, 256, 0, stream>>>(acts, labels,
                                                         lp_blank, lp_emit);
    rnnt_alpha_kernel<<<B_, 64, 0, stream>>>(lp_blank, lp_emit,
                                             act_lens, label_lens, out);
}